// DecoderWithAttention_20864951124378
// MI455X (gfx1250) — compile-verified
//
#include <hip/hip_runtime.h>
#include <math.h>

// ---------------------------------------------------------------------------
// Problem sizes (from reference): V=32000, E=300, H=1024, BS=256, S=512
// Outputs (flat, in order): logits (256*32000), h_new (256*1024), attn (256*512)
// ---------------------------------------------------------------------------
#define V_SZ 32000
#define E_SZ 300
#define H_SZ 1024
#define BS_SZ 256
#define S_SZ 512

typedef __attribute__((ext_vector_type(2))) float v2f;
typedef __attribute__((ext_vector_type(8))) float v8f;

// ---------------------------------------------------------------------------
// Kernel A: fused attention with online (flash-style) softmax.
// One block per batch row b. 8 waves; each wave owns rows s = wave + 8*i.
// Per-lane ctx accumulator: 8 float4 chunks => dims d = 4*lane + 128*j.
// Single pass over encoder_outputs (512 MB read once @ 23.3 TB/s).
// ---------------------------------------------------------------------------
__global__ void __launch_bounds__(256) attn_kernel(
    const float* __restrict__ enc,     // (BS, S, H)
    const float* __restrict__ hprev,   // (BS, H)
    float* __restrict__ ctx,           // (BS, H) workspace
    float* __restrict__ attn)          // (BS, S) output
{
    const int tid  = threadIdx.x;
    const int lane = tid & 31;
    const int wave = tid >> 5;
    const int b    = blockIdx.x;

    __shared__ float  s_energy[S_SZ];
    __shared__ float  s_m[8], s_l[8];
    __shared__ float4 s_acc[8][256];   // 32 KB

    // Load this lane's slice of h[b] (32 floats as 8 float4)
    const float4* hrow = (const float4*)(hprev + (size_t)b * H_SZ);
    float4 hv[8];
#pragma unroll
    for (int j = 0; j < 8; ++j) hv[j] = hrow[lane + 32 * j];

    float4 acc[8];
#pragma unroll
    for (int j = 0; j < 8; ++j) acc[j] = make_float4(0.f, 0.f, 0.f, 0.f);
    float m = -INFINITY;
    float l = 0.f;

    for (int s = wave; s < S_SZ; s += 8) {
        const float4* row = (const float4*)(enc + ((size_t)b * S_SZ + s) * H_SZ);
        float4 r[8];
        float  e = 0.f;
#pragma unroll
        for (int j = 0; j < 8; ++j) {
            r[j] = row[lane + 32 * j];
            e += r[j].x * hv[j].x + r[j].y * hv[j].y +
                 r[j].z * hv[j].z + r[j].w * hv[j].w;
        }
        // wave32 all-reduce of the dot product
#pragma unroll
        for (int off = 16; off > 0; off >>= 1) e += __shfl_xor(e, off, 32);

        if (lane == 0) s_energy[s] = e;

        const float mn    = fmaxf(m, e);
        const float scale = __expf(m - mn);   // first iter: exp(-inf)=0
        const float p     = __expf(e - mn);
        l = l * scale + p;
#pragma unroll
        for (int j = 0; j < 8; ++j) {
            acc[j].x = acc[j].x * scale + p * r[j].x;
            acc[j].y = acc[j].y * scale + p * r[j].y;
            acc[j].z = acc[j].z * scale + p * r[j].z;
            acc[j].w = acc[j].w * scale + p * r[j].w;
        }
        m = mn;
    }

    if (lane == 0) { s_m[wave] = m; s_l[wave] = l; }
#pragma unroll
    for (int j = 0; j < 8; ++j) s_acc[wave][lane + 32 * j] = acc[j];
    __syncthreads();

    // Combine the 8 waves' partial softmax states
    float M = s_m[0];
#pragma unroll
    for (int w = 1; w < 8; ++w) M = fmaxf(M, s_m[w]);
    float L = 0.f;
#pragma unroll
    for (int w = 0; w < 8; ++w) L += s_l[w] * __expf(s_m[w] - M);
    const float invL = 1.0f / L;

    // ctx: thread tid reduces chunk index tid across waves
    float4 sum = make_float4(0.f, 0.f, 0.f, 0.f);
#pragma unroll
    for (int w = 0; w < 8; ++w) {
        const float  f = __expf(s_m[w] - M);
        const float4 a = s_acc[w][tid];
        sum.x += f * a.x; sum.y += f * a.y; sum.z += f * a.z; sum.w += f * a.w;
    }
    sum.x *= invL; sum.y *= invL; sum.z *= invL; sum.w *= invL;
    const int dimbase = 4 * (tid & 31) + 128 * (tid >> 5);
    *(float4*)(ctx + (size_t)b * H_SZ + dimbase) = sum;

    // attention weights output
    for (int s = tid; s < S_SZ; s += 256)
        attn[(size_t)b * S_SZ + s] = __expf(s_energy[s] - M) * invL;
}

// ---------------------------------------------------------------------------
// WMMA helper: f32 16x16x4, D = A x B + C
// A layout (16x4): lane l holds row m = l%16; VGPR0 = K=(l>=16?2:0), VGPR1 = K+1
// B layout (4x16): lane l holds col n = l%16; same K split.
// C/D (16x16): lane l col n = l%16; VGPR v holds row m = v + 8*(l>=16).
// ---------------------------------------------------------------------------
static __device__ __forceinline__ v8f wmma_f32(v2f a, v2f b, v8f c) {
    return __builtin_amdgcn_wmma_f32_16x16x4_f32(
        /*neg_a=*/false, a, /*neg_b=*/false, b,
        /*c_mod=*/(short)0, c, /*reuse_a=*/false, /*reuse_b=*/false);
}

// ---------------------------------------------------------------------------
// Kernel B: h_new = tanh( [emb, ctx] @ W_ih^T + b_ih + h @ W_hh^T + b_hh )
// 1024 tiles (16 m-tiles x 64 n-tiles), one wave per tile, 8 waves/block.
// ---------------------------------------------------------------------------
__global__ void __launch_bounds__(256) rnn_kernel(
    const int*   __restrict__ idx,       // (BS,)
    const float* __restrict__ hprev,     // (BS, H)
    const float* __restrict__ ctx,       // (BS, H)
    const float* __restrict__ emb_table, // (V, E)
    const float* __restrict__ W_ih,      // (H, E+H)
    const float* __restrict__ b_ih,      // (H,)
    const float* __restrict__ W_hh,      // (H, H)
    const float* __restrict__ b_hh,      // (H,)
    float* __restrict__ h_new)           // (BS, H)
{
    const int tid  = threadIdx.x;
    const int lane = tid & 31;
    const int wave = tid >> 5;
    const int t    = blockIdx.x * 8 + wave;     // 0..1023
    const int m_tile = t & 15;
    const int n_tile = t >> 4;

    const int mr   = lane & 15;
    const int half = lane >> 4;
    const int koff = half * 2;

    const int m_g = m_tile * 16 + mr;   // A row (batch)
    const int n_g = n_tile * 16 + mr;   // B col (hidden unit)

    const float* embrow = emb_table + (size_t)idx[m_g] * E_SZ + koff;
    const float* ctxrow = ctx   + (size_t)m_g * H_SZ + koff;
    const float* hrow   = hprev + (size_t)m_g * H_SZ + koff;
    const float* wih    = W_ih  + (size_t)n_g * (E_SZ + H_SZ) + koff;
    const float* whh    = W_hh  + (size_t)n_g * H_SZ + koff;

    v8f c = (v8f)(0.0f);

    // Phase 1: embedding part of x (K = 0..300, 300 % 4 == 0)
    for (int k = 0; k < E_SZ; k += 4) {
        v2f a  = *(const v2f*)(embrow + k);
        v2f bb = *(const v2f*)(wih + k);
        c = wmma_f32(a, bb, c);
    }
    // Phase 2: ctx part of x (K = 300..1324)
    for (int k = 0; k < H_SZ; k += 4) {
        v2f a  = *(const v2f*)(ctxrow + k);
        v2f bb = *(const v2f*)(wih + E_SZ + k);
        c = wmma_f32(a, bb, c);
    }
    // Phase 3: recurrent term h @ W_hh^T
    for (int k = 0; k < H_SZ; k += 4) {
        v2f a  = *(const v2f*)(hrow + k);
        v2f bb = *(const v2f*)(whh + k);
        c = wmma_f32(a, bb, c);
    }

    const float bias = b_ih[n_g] + b_hh[n_g];
#pragma unroll
    for (int v = 0; v < 8; ++v) {
        const int m_out = m_tile * 16 + v + half * 8;
        h_new[(size_t)m_out * H_SZ + n_g] = tanhf(c[v] + bias);
    }
}

// ---------------------------------------------------------------------------
// Kernel C: logits = h_new @ fc_W^T + fc_b   (256 x 32000, K=1024)
// One block per 16-wide n-tile (2000 blocks); 16 waves cover all 16 m-tiles,
// so the shared fc_W fragments hit WGP$/L2 and fc_W streams from HBM once.
// ---------------------------------------------------------------------------
__global__ void __launch_bounds__(512) logits_kernel(
    const float* __restrict__ h_new,   // (BS, H)
    const float* __restrict__ fc_W,    // (V, H)
    const float* __restrict__ fc_b,    // (V,)
    float* __restrict__ logits)        // (BS, V)
{
    const int tid  = threadIdx.x;
    const int lane = tid & 31;
    const int m_tile = tid >> 5;        // wave id = m-tile, 0..15
    const int n_tile = blockIdx.x;      // 0..1999

    const int mr   = lane & 15;
    const int half = lane >> 4;
    const int koff = half * 2;

    const int m_g = m_tile * 16 + mr;
    const int n_g = n_tile * 16 + mr;

    const float* arow = h_new + (size_t)m_g * H_SZ + koff;
    const float* brow = fc_W  + (size_t)n_g * H_SZ + koff;

    v8f c = (v8f)(0.0f);
    for (int k = 0; k < H_SZ; k += 4) {
        v2f a  = *(const v2f*)(arow + k);
        v2f bb = *(const v2f*)(brow + k);
        c = wmma_f32(a, bb, c);
    }

    const float bias = fc_b[n_g];
#pragma unroll
    for (int v = 0; v < 8; ++v) {
        const int m_out = m_tile * 16 + v + half * 8;
        logits[(size_t)m_out * V_SZ + n_g] = c[v] + bias;
    }
}

// ---------------------------------------------------------------------------
// Launch
// ---------------------------------------------------------------------------
extern "C" void kernel_launch(void* const* d_in, const int* in_sizes, int n_in,
                              void* d_out, int out_size, void* d_ws, size_t ws_size,
                              hipStream_t stream) {
    const int*   idx   = (const int*)  d_in[0];
    const float* hprev = (const float*)d_in[1];   // (1, BS, H)
    const float* enc   = (const float*)d_in[2];   // (BS, S, H)
    const float* emb   = (const float*)d_in[3];   // (V, E)
    const float* W_ih  = (const float*)d_in[4];
    const float* b_ih  = (const float*)d_in[5];
    const float* W_hh  = (const float*)d_in[6];
    const float* b_hh  = (const float*)d_in[7];
    const float* fc_W  = (const float*)d_in[8];
    const float* fc_b  = (const float*)d_in[9];

    float* out    = (float*)d_out;
    float* logits = out;                                   // 256*32000
    float* h_new  = out + (size_t)BS_SZ * V_SZ;            // 256*1024
    float* attn   = h_new + (size_t)BS_SZ * H_SZ;          // 256*512

    float* ctx = (float*)d_ws;                             // 256*1024 f32

    attn_kernel  <<<BS_SZ, 256, 0, stream>>>(enc, hprev, ctx, attn);
    rnn_kernel   <<<(16 * 64) / 8, 256, 0, stream>>>(idx, hprev, ctx, emb,
                                                     W_ih, b_ih, W_hh, b_hh, h_new);
    logits_kernel<<<V_SZ / 16, 512, 0, stream>>>(h_new, fc_W, fc_b, logits);
}